// ROISelect_38534446579958
// MI455X (gfx1250) — compile-verified
//
#include <hip/hip_runtime.h>
#include <stdint.h>

// Problem constants (from reference): B=128, N=131072, K=256
#define N_DIM 131072
#define K_DIM 256
#define NBINS 4096          // top-12 bits of order-transformed float
#define NCHUNK (NBINS / 32)
#define CAP   4096          // candidate buffer capacity (64-bit entries)
#define NTHREADS 1024       // 32 wave32 waves per workgroup
#define TILE_FLOATS (NTHREADS * 4)       // 4096 floats = 16KB per tile buffer
#define NTILES (N_DIM / TILE_FLOATS)     // 32 tiles per row
#define NBUF 8                            // async depth: 128KB in flight per WGP

// Map float -> uint32 such that unsigned compare == float descending-order compare
__device__ __forceinline__ unsigned key_xform(float f) {
    unsigned u = __float_as_uint(f);
    return (u & 0x80000000u) ? ~u : (u | 0x80000000u);
}
__device__ __forceinline__ float key_unxform(unsigned u) {
    unsigned orig = (u & 0x80000000u) ? (u ^ 0x80000000u) : ~u;
    return __uint_as_float(orig);
}

__global__ __launch_bounds__(NTHREADS, 1)
void ROISelect_topk_kernel(const float* __restrict__ score,
                           const float* __restrict__ roi,
                           float* __restrict__ out_roi,     // [B, K, 4]
                           float* __restrict__ out_score) { // [B, K]
    __shared__ unsigned hist[NBINS];
    __shared__ unsigned chunkSum[NCHUNK];
    // Tile staging (pass 1) and candidate/sort buffer (pass 2+) never coexist:
    // alias them. Union = max(8*16KB, 32KB) = 128KB; total ~144.6KB (<320KB/WGP).
    __shared__ union SharedU {
        __align__(16) float tiles[NBUF][TILE_FLOATS];    // 128KB async staging
        unsigned long long cand[CAP];                    // 32KB sort keys
    } u;
    __shared__ unsigned sThreshBin;
    __shared__ unsigned sCount;
    __shared__ unsigned sS;

    const int b   = blockIdx.x;
    const int tid = threadIdx.x;
    const float* srow = score + (size_t)b * N_DIM;
    const int NV = N_DIM / 4; // 32768 float4 per row

    for (int i = tid; i < NBINS; i += NTHREADS) hist[i] = 0u;
    if (tid == 0) sCount = 0u;
    __syncthreads();

    // ---- Pass 1: histogram, streaming score via async global->LDS copies ----
    // Depth-8 pipeline: issue tile t+7 while histogramming tile t. Each lane owns
    // a private 16B slot per buffer (no cross-lane sharing -> no barriers).
    // ASYNCcnt ops complete in order: after issuing t+7, cnt<=7 => tile t landed.
    // In flight: 8 ops x 512B x 32 waves = 128KB/WGP; x128 WGs = 16MB chip-wide
    // = 23.3TB/s x ~700ns, i.e. the HBM latency-bandwidth product.
    {
        unsigned ldsOff[NBUF];
        #pragma unroll
        for (int q = 0; q < NBUF; ++q)
            ldsOff[q] = (unsigned)(uintptr_t)&u.tiles[q][tid * 4];
        const uint64_t g0 = (uint64_t)(uintptr_t)srow + (uint64_t)tid * 16u;

        #pragma unroll
        for (int p = 0; p < NBUF - 1; ++p) {  // prologue: tiles 0..NBUF-2 in flight
            uint64_t ga = g0 + (uint64_t)p * (TILE_FLOATS * 4u);
            asm volatile("global_load_async_to_lds_b128 %0, %1, off"
                         :: "v"(ldsOff[p]), "v"(ga) : "memory");
        }

#define CONSUME_TILE(t_) do {                                                     \
        const float4 v_ = *reinterpret_cast<const float4*>(                       \
            &u.tiles[(t_) & (NBUF - 1)][tid * 4]);                                \
        atomicAdd(&hist[key_xform(v_.x) >> 20], 1u);                              \
        atomicAdd(&hist[key_xform(v_.y) >> 20], 1u);                              \
        atomicAdd(&hist[key_xform(v_.z) >> 20], 1u);                              \
        atomicAdd(&hist[key_xform(v_.w) >> 20], 1u);                              \
    } while (0)

        // Body: iterations that still issue a new tile.
        const int BODY = NTILES - NBUF + 1;
        for (int t = 0; t < BODY; ++t) {
            uint64_t ga = g0 + (uint64_t)(t + NBUF - 1) * (TILE_FLOATS * 4u);
            asm volatile("global_load_async_to_lds_b128 %0, %1, off"
                         :: "v"(ldsOff[(t + NBUF - 1) & (NBUF - 1)]), "v"(ga)
                         : "memory");
            asm volatile("s_wait_asynccnt 0x7" ::: "memory");
            CONSUME_TILE(t);
        }
        // Pipelined drain: wait on exactly (NTILES-1-t) outstanding ops so the
        // remaining tiles retire without serializing (constant immediates via
        // full unroll of a constant-trip loop).
        #pragma unroll
        for (int t = BODY; t < NTILES; ++t) {
            switch (NTILES - 1 - t) {
            case 6: asm volatile("s_wait_asynccnt 0x6" ::: "memory"); break;
            case 5: asm volatile("s_wait_asynccnt 0x5" ::: "memory"); break;
            case 4: asm volatile("s_wait_asynccnt 0x4" ::: "memory"); break;
            case 3: asm volatile("s_wait_asynccnt 0x3" ::: "memory"); break;
            case 2: asm volatile("s_wait_asynccnt 0x2" ::: "memory"); break;
            case 1: asm volatile("s_wait_asynccnt 0x1" ::: "memory"); break;
            default: asm volatile("s_wait_asynccnt 0x0" ::: "memory"); break;
            }
            CONSUME_TILE(t);
        }
#undef CONSUME_TILE
    }
    __syncthreads();

    // ---- Find threshold bin: largest bin t with count(bins >= t) >= K ----
    if (tid < NCHUNK) {
        unsigned s = 0;
        #pragma unroll
        for (int j = 0; j < 32; ++j) s += hist[tid * 32 + j];
        chunkSum[tid] = s;
    }
    __syncthreads();

    if (tid == 0) {
        unsigned cum = 0, tbin = 0;
        for (int c = NCHUNK - 1; c >= 0; --c) {
            if (cum + chunkSum[c] >= K_DIM) {
                for (int bin = c * 32 + 31; bin >= c * 32; --bin) {
                    if (cum + hist[bin] >= K_DIM) { tbin = (unsigned)bin; break; }
                    cum += hist[bin];
                }
                break;
            }
            cum += chunkSum[c];
        }
        sThreshBin = tbin;
    }
    __syncthreads();
    const unsigned tbin = sThreshBin;

    // ---- Pass 2: collect candidates with bin >= tbin (re-read hits 192MB L2) ----
    // Unrolled x4 for MLP: 4 independent b128 loads in flight per wave.
    const float4* __restrict__ srow4 = reinterpret_cast<const float4*>(srow);
#define COLLECT(val, idx_) do {                                                   \
        unsigned key_ = key_xform(val);                                           \
        if ((key_ >> 20) >= tbin) {                                               \
            unsigned pos_ = atomicAdd(&sCount, 1u); /* ds_add_rtn_u32 */          \
            if (pos_ < CAP)                                                       \
                u.cand[pos_] = ((unsigned long long)key_ << 32) |                 \
                               (unsigned)(~(idx_));                               \
        } } while (0)

    for (int i = tid; i < NV; i += NTHREADS * 4) {  // NV % (4*NTHREADS) == 0
        float4 v0 = srow4[i];
        float4 v1 = srow4[i + NTHREADS];
        float4 v2 = srow4[i + 2 * NTHREADS];
        float4 v3 = srow4[i + 3 * NTHREADS];
        __builtin_prefetch(&srow4[i + 4 * NTHREADS], 0, 1); // global_prefetch_b8
        COLLECT(v0.x, (unsigned)(i) * 4u + 0u);
        COLLECT(v0.y, (unsigned)(i) * 4u + 1u);
        COLLECT(v0.z, (unsigned)(i) * 4u + 2u);
        COLLECT(v0.w, (unsigned)(i) * 4u + 3u);
        COLLECT(v1.x, (unsigned)(i + NTHREADS) * 4u + 0u);
        COLLECT(v1.y, (unsigned)(i + NTHREADS) * 4u + 1u);
        COLLECT(v1.z, (unsigned)(i + NTHREADS) * 4u + 2u);
        COLLECT(v1.w, (unsigned)(i + NTHREADS) * 4u + 3u);
        COLLECT(v2.x, (unsigned)(i + 2 * NTHREADS) * 4u + 0u);
        COLLECT(v2.y, (unsigned)(i + 2 * NTHREADS) * 4u + 1u);
        COLLECT(v2.z, (unsigned)(i + 2 * NTHREADS) * 4u + 2u);
        COLLECT(v2.w, (unsigned)(i + 2 * NTHREADS) * 4u + 3u);
        COLLECT(v3.x, (unsigned)(i + 3 * NTHREADS) * 4u + 0u);
        COLLECT(v3.y, (unsigned)(i + 3 * NTHREADS) * 4u + 1u);
        COLLECT(v3.z, (unsigned)(i + 3 * NTHREADS) * 4u + 2u);
        COLLECT(v3.w, (unsigned)(i + 3 * NTHREADS) * 4u + 3u);
    }
#undef COLLECT
    __syncthreads();

    unsigned C = sCount; if (C > CAP) C = CAP; // C >= K by construction
    if (tid == 0) {
        unsigned S = K_DIM;
        while (S < C) S <<= 1;
        sS = S;
    }
    __syncthreads();
    const unsigned S = sS;
    for (unsigned i = C + tid; i < S; i += NTHREADS) u.cand[i] = 0ull; // pad: -inf key
    __syncthreads();

    // ---- Bitonic sort descending over S (<= 4096) packed 64-bit keys ----
    for (unsigned k = 2; k <= S; k <<= 1) {
        for (unsigned j = k >> 1; j > 0; j >>= 1) {
            for (unsigned i = tid; i < S; i += NTHREADS) {
                unsigned ixj = i ^ j;
                if (ixj > i) {
                    unsigned long long a  = u.cand[i];
                    unsigned long long c2 = u.cand[ixj];
                    bool desc = ((i & k) == 0);
                    if (desc ? (a < c2) : (a > c2)) { u.cand[i] = c2; u.cand[ixj] = a; }
                }
            }
            __syncthreads();
        }
    }

    // ---- Emit top-K scores and gather ROI rows (float4) ----
    if (tid < K_DIM) {
        unsigned long long e = u.cand[tid];
        unsigned key = (unsigned)(e >> 32);
        unsigned idx = ~(unsigned)e;
        out_score[(size_t)b * K_DIM + tid] = key_unxform(key);
        const float4* r4 = reinterpret_cast<const float4*>(roi) + (size_t)b * N_DIM + idx;
        reinterpret_cast<float4*>(out_roi)[(size_t)b * K_DIM + tid] = *r4;
    }
}

extern "C" void kernel_launch(void* const* d_in, const int* in_sizes, int n_in,
                              void* d_out, int out_size, void* d_ws, size_t ws_size,
                              hipStream_t stream) {
    const float* score = (const float*)d_in[0];   // [B, N] f32
    const float* roi   = (const float*)d_in[1];   // [B, N, 4] f32
    // d_in[2] is k (==256), compiled in as K_DIM

    const int B = in_sizes[0] / N_DIM;            // 128
    float* out_roi   = (float*)d_out;             // [B, K, 4] first
    float* out_score = (float*)d_out + (size_t)B * K_DIM * 4; // [B, K] second

    ROISelect_topk_kernel<<<B, NTHREADS, 0, stream>>>(score, roi, out_roi, out_score);
}